// Mismatch_loss_19018115187333
// MI455X (gfx1250) — compile-verified
//
#include <hip/hip_runtime.h>

// Problem geometry: B=8, C=16, H=W=512  ->  128 planes of 262144 fp32 elements.
#define PLANES   128
#define HW       262144
#define HW4      65536          // HW / 4 (float4 elements per plane)
#define SPLIT    8              // blocks per plane
#define TPB      256            // 8 waves of 32
#define CHUNK4   (HW4 / SPLIT)  // 8192 float4 per block, 32 iters/thread

typedef __attribute__((ext_vector_type(2))) float v2f;
typedef __attribute__((ext_vector_type(8))) float v8f;

// ---------------------------------------------------------------------------
// Intra-wave (32-lane) exact fp32 sum using V_WMMA_F32_16X16X4_F32.
// A = all-ones 16x4. Then D[m][n] = sum over K of B[k][n]; column n of B is
// fed by lanes {n, n+16} x {VGPR0, VGPR1} regardless of the exact K<->VGPR
// mapping, so with b = {x[l], x[l^8]} every column sums a disjoint group of
// 4 lanes of x; three xor-folds finish the 32-lane sum. All adds are fp32 RNE.
// ---------------------------------------------------------------------------
__device__ __forceinline__ float wave_sum32(float x) {
  float y = __shfl_xor(x, 8, 32);
  v2f a; a[0] = 1.0f; a[1] = 1.0f;
  v2f b; b[0] = x;    b[1] = y;
  v8f c = {};
  c = __builtin_amdgcn_wmma_f32_16x16x4_f32(
      /*neg_a=*/false, a, /*neg_b=*/false, b,
      /*c_mod=*/(short)0, c, /*reuse_a=*/false, /*reuse_b=*/false);
  float s = c[0];
  s += __shfl_xor(s, 1, 32);
  s += __shfl_xor(s, 2, 32);
  s += __shfl_xor(s, 4, 32);
  return s;   // every lane holds the full 32-lane sum
}

// ---------------------------------------------------------------------------
// Kernel 0: zero the 128*4-float accumulator region of the workspace.
// ---------------------------------------------------------------------------
__global__ void zero_ws_kernel(float* __restrict__ ws) {
  int i = threadIdx.x;
  if (i < PLANES * 4) ws[i] = 0.0f;
}

// ---------------------------------------------------------------------------
// Kernel 1: bandwidth phase. One block handles 1/8 of a plane with float4
// (global_load_b128) streaming loads. Accumulates 4 sums per thread, reduces
// per-wave via WMMA, per-block via LDS, per-plane via global f32 atomics.
// ---------------------------------------------------------------------------
__global__ __launch_bounds__(TPB) void plane_partials_kernel(
    const float* __restrict__ net, const float* __restrict__ tgt,
    const float* __restrict__ mpo, float* __restrict__ ws) {
  const int plane = blockIdx.x / SPLIT;
  const int chunk = blockIdx.x % SPLIT;
  const size_t base4 = (size_t)plane * HW4 + (size_t)chunk * CHUNK4;

  const float4* __restrict__ n4 = (const float4*)net + base4;
  const float4* __restrict__ t4 = (const float4*)tgt + base4;
  const float4* __restrict__ m4 = (const float4*)mpo + base4;

  float acc_tn = 0.0f, acc_t = 0.0f, acc_n = 0.0f, acc_m = 0.0f;

  for (int i = threadIdx.x; i < CHUNK4; i += TPB) {   // uniform trip count
    float4 n = n4[i];
    float4 t = t4[i];
    float4 m = m4[i];
    acc_tn = fmaf(t.x, n.x, acc_tn);
    acc_tn = fmaf(t.y, n.y, acc_tn);
    acc_tn = fmaf(t.z, n.z, acc_tn);
    acc_tn = fmaf(t.w, n.w, acc_tn);
    acc_t += (t.x + t.y) + (t.z + t.w);
    acc_n += (n.x + n.y) + (n.z + n.w);
    acc_m += (m.x + m.y) + (m.z + m.w);
  }

  // EXEC is all-1s here (no divergence yet): legal to run WMMA.
  float s_tn = wave_sum32(acc_tn);
  float s_t  = wave_sum32(acc_t);
  float s_n  = wave_sum32(acc_n);
  float s_m  = wave_sum32(acc_m);

  __shared__ float red[TPB / 32][4];
  const int wave = threadIdx.x >> 5;
  const int lane = threadIdx.x & 31;
  if (lane == 0) {
    red[wave][0] = s_tn; red[wave][1] = s_t;
    red[wave][2] = s_n;  red[wave][3] = s_m;
  }
  __syncthreads();
  if (threadIdx.x < 4) {
    float s = 0.0f;
#pragma unroll
    for (int w = 0; w < TPB / 32; ++w) s += red[w][threadIdx.x];
    atomicAdd(&ws[plane * 4 + threadIdx.x], s);
  }
}

// ---------------------------------------------------------------------------
// Kernel 2: [8,16] epilogue -> scalar. One block, 128 threads (one per plane).
// ---------------------------------------------------------------------------
__global__ void finalize_kernel(const float* __restrict__ ws,
                                float* __restrict__ out) {
  __shared__ float loss_s[PLANES];
  __shared__ float img_s[8];
  const int p = threadIdx.x;            // plane = b*16 + c
  const float b2 = 1.5f * 1.5f;
  const float w1 = b2 / (1.0f + b2);
  const float w2 = 1.0f / (1.0f + b2);

  float s_tn = ws[4 * p + 0];
  float s_t  = ws[4 * p + 1];
  float s_n  = ws[4 * p + 2];
  float s_m  = ws[4 * p + 3];

  float mol  = s_tn;
  float fn   = s_t - s_tn;              // sum t*(1-n)
  float fp   = s_n - s_tn;              // sum (1-t)*n
  float loss = 1.0f - mol / (mol + w1 * fn + w2 * fp);
  // masks are exactly {0,1}: max>0  <=>  sum>0 (sums exact: < 2^24 integers)
  bool  active = (s_t > 0.0f) || (s_m > 0.0f);
  loss_s[p] = active ? loss : 0.0f;
  __syncthreads();

  if (p < 8) {                          // per-image: sum / count_nonzero
    float sum = 0.0f, cnt = 0.0f;
#pragma unroll
    for (int c = 0; c < 16; ++c) {
      float v = loss_s[p * 16 + c];
      sum += v;
      cnt += (v != 0.0f) ? 1.0f : 0.0f;
    }
    img_s[p] = sum / cnt;               // 0/0 -> NaN, matching reference
  }
  __syncthreads();

  if (p == 0) {
    float s = 0.0f;
#pragma unroll
    for (int b = 0; b < 8; ++b) s += img_s[b];
    out[0] = s * 0.125f;
  }
}

// ---------------------------------------------------------------------------
extern "C" void kernel_launch(void* const* d_in, const int* in_sizes, int n_in,
                              void* d_out, int out_size, void* d_ws, size_t ws_size,
                              hipStream_t stream) {
  const float* net = (const float*)d_in[0];   // net_out        [8,16,512,512] f32
  const float* tgt = (const float*)d_in[1];   // target         [8,16,512,512] f32
  const float* mpo = (const float*)d_in[2];   // max_positiones [8,16,512,512] f32
  float* out = (float*)d_out;                 // scalar f32
  float* ws  = (float*)d_ws;                  // >= 128*4 floats

  zero_ws_kernel<<<1, 512, 0, stream>>>(ws);
  plane_partials_kernel<<<PLANES * SPLIT, TPB, 0, stream>>>(net, tgt, mpo, ws);
  finalize_kernel<<<1, PLANES, 0, stream>>>(ws, out);
}